// SparseMultiHeadAttention_86758339379875
// MI455X (gfx1250) — compile-verified
//
#include <hip/hip_runtime.h>
#include <hip/hip_bf16.h>

// ---------------------------------------------------------------------------
// SparseMultiHeadAttention for MI455X (gfx1250, wave32)
// f32 end-to-end using V_WMMA_F32_16X16X4_F32 (memory-bound problem,
// full-precision matrix path is free vs the 23.3 TB/s HBM roofline).
// Tile staging uses GLOBAL_LOAD_ASYNC_TO_LDS_B128 (ASYNCcnt) when available.
// ---------------------------------------------------------------------------

#define B_  8
#define S_  2048
#define D_  1024
#define H_  16
#define A_  4
#define DA_ 64
#define LN_EPS 1e-5f

typedef float v2f __attribute__((ext_vector_type(2)));
typedef float v8f __attribute__((ext_vector_type(8)));
typedef int   v4i __attribute__((ext_vector_type(4)));

__device__ __forceinline__ v8f wmma4(v2f a, v2f b, v8f c) {
  // D = A(16x4 f32) * B(4x16 f32) + C(16x16 f32)
  return __builtin_amdgcn_wmma_f32_16x16x4_f32(
      /*neg_a=*/false, a, /*neg_b=*/false, b,
      /*c_mod=*/(short)0, c, /*reuse_a=*/false, /*reuse_b=*/false);
}

// --------------------- async global -> LDS staging -------------------------
#if defined(__gfx1250__) && __has_builtin(__builtin_amdgcn_global_load_async_to_lds_b128)
#define HAVE_ASYNC_LDS 1
#else
#define HAVE_ASYNC_LDS 0
#endif

typedef __attribute__((address_space(1))) v4i* gptr_b128;
typedef __attribute__((address_space(3))) v4i* lptr_b128;

__device__ __forceinline__ void cp_b128(float* lds, const float* g) {
#if HAVE_ASYNC_LDS
  __builtin_amdgcn_global_load_async_to_lds_b128(
      (gptr_b128)(g), (lptr_b128)(lds), /*imm offset=*/0, /*cpol=*/0);
#else
  *(float4*)lds = *(const float4*)g;
#endif
}

__device__ __forceinline__ void cp_wait() {
#if HAVE_ASYNC_LDS
#if __has_builtin(__builtin_amdgcn_s_wait_asynccnt)
  __builtin_amdgcn_s_wait_asynccnt(0);
#else
  asm volatile("s_wait_asynccnt 0" ::: "memory");
#endif
#endif
}

// Workspace layout (float-element offsets)
static constexpr size_t XBAR_OFF = 0;                     // B*D = 8192 f32
static constexpr size_t GATE_OFF = 8192;                  // B*A = 32 f32
static constexpr size_t IDX_OFF  = 8256;                  // B*A = 32 i32
static constexpr size_t QACT_OFF = 16384;                 // B*A*S*DA f32
static constexpr size_t ACT_SZ   = (size_t)B_ * A_ * S_ * DA_;   // 4,194,304
static constexpr size_t KACT_OFF = QACT_OFF + ACT_SZ;
static constexpr size_t VACT_OFF = KACT_OFF + ACT_SZ;
static constexpr size_t OFIN_OFF = VACT_OFF + ACT_SZ;     // B*S*(A*DA) f32

// ---------------------------------------------------------------------------
// Kernel 1: xbar[b][d] = mean_s x[b][s][d]   (mean commutes with x @ Wr)
// grid = B * (D/256), block = 256
// ---------------------------------------------------------------------------
__global__ __launch_bounds__(256) void mean_x_kernel(const float* __restrict__ x,
                                                     float* __restrict__ xbar) {
  int b    = blockIdx.x / (D_ / 256);
  int dblk = blockIdx.x % (D_ / 256);
  int d    = dblk * 256 + threadIdx.x;
  const float* xb = x + (size_t)b * S_ * D_;
  float s = 0.0f;
  for (int t = 0; t < S_; ++t) {
    __builtin_prefetch(xb + (size_t)(t + 16) * D_ + d, 0, 1);
    s += xb[(size_t)t * D_ + d];
  }
  xbar[b * D_ + d] = s * (1.0f / (float)S_);
}

// ---------------------------------------------------------------------------
// Kernel 2: routing — r = xbar @ Wr + br; LN over H; softmax; top-4 (sorted)
// grid = B, block = 256 (16 partial dots per head, reduce, then serial tail)
// ---------------------------------------------------------------------------
__global__ __launch_bounds__(256) void route_kernel(const float* __restrict__ xbar,
                                                    const float* __restrict__ Wr,
                                                    const float* __restrict__ br,
                                                    const float* __restrict__ gamma,
                                                    const float* __restrict__ beta,
                                                    int* __restrict__ idx_out,
                                                    float* __restrict__ gate_out) {
  __shared__ float part[16][17];
  int b = blockIdx.x;
  int h = threadIdx.x & 15;
  int p = threadIdx.x >> 4;   // 0..15
  float s = 0.0f;
  for (int k = p * 64; k < p * 64 + 64; ++k)
    s += xbar[b * D_ + k] * Wr[k * H_ + h];
  part[p][h] = s;
  __syncthreads();
  if (threadIdx.x == 0) {
    float r[H_];
    for (int hh = 0; hh < H_; ++hh) {
      float acc = br[hh];
      for (int pp = 0; pp < 16; ++pp) acc += part[pp][hh];
      r[hh] = acc;
    }
    float mu = 0.0f;
    for (int hh = 0; hh < H_; ++hh) mu += r[hh];
    mu *= (1.0f / H_);
    float var = 0.0f;
    for (int hh = 0; hh < H_; ++hh) { float d = r[hh] - mu; var += d * d; }
    var *= (1.0f / H_);
    float inv = rsqrtf(var + LN_EPS);
    float rn[H_];
    float mx = -3.0e38f;
    for (int hh = 0; hh < H_; ++hh) {
      rn[hh] = (r[hh] - mu) * inv * gamma[hh] + beta[hh];
      mx = fmaxf(mx, rn[hh]);
    }
    float dist[H_];
    float sum = 0.0f;
    for (int hh = 0; hh < H_; ++hh) { dist[hh] = __expf(rn[hh] - mx); sum += dist[hh]; }
    float rs = 1.0f / sum;
    for (int hh = 0; hh < H_; ++hh) dist[hh] *= rs;
    // top-4, first-occurrence on ties (matches jax.lax.top_k), then sort idx
    int sel[A_];
    bool used[H_];
    for (int hh = 0; hh < H_; ++hh) used[hh] = false;
    for (int a = 0; a < A_; ++a) {
      int best = -1; float bv = -1.0f;
      for (int hh = 0; hh < H_; ++hh)
        if (!used[hh] && dist[hh] > bv) { bv = dist[hh]; best = hh; }
      used[best] = true;
      sel[a] = best;
    }
    for (int i = 1; i < A_; ++i) {            // insertion sort ascending
      int v = sel[i], j = i - 1;
      while (j >= 0 && sel[j] > v) { sel[j + 1] = sel[j]; --j; }
      sel[j + 1] = v;
    }
    for (int a = 0; a < A_; ++a) {
      idx_out[b * A_ + a]  = sel[a];
      gate_out[b * A_ + a] = dist[sel[a]];
    }
  }
}

// ---------------------------------------------------------------------------
// Kernel 3: gathered-head QKV projection.
// out[b][a][s][:] = x_b(2048x1024) @ W[:, h*64 : h*64+64] + bias[h*64:...]
// grid = (S/128, B*A, 3), block = 256 (8 waves). Block tile M=128, N=64.
// ---------------------------------------------------------------------------
__global__ __launch_bounds__(256) void qkv_proj_kernel(
    const float* __restrict__ x,
    const float* __restrict__ Wq, const float* __restrict__ bq,
    const float* __restrict__ Wk, const float* __restrict__ bk,
    const float* __restrict__ Wv, const float* __restrict__ bv,
    const int* __restrict__ idx,
    float* __restrict__ Qa, float* __restrict__ Ka, float* __restrict__ Va) {
  const float* W    = (blockIdx.z == 0) ? Wq : (blockIdx.z == 1) ? Wk : Wv;
  const float* bias = (blockIdx.z == 0) ? bq : (blockIdx.z == 1) ? bk : bv;
  float*       out  = (blockIdx.z == 0) ? Qa : (blockIdx.z == 1) ? Ka : Va;

  __shared__ float xs[128][36];   // 128 rows x 32 K (padded stride 36)
  __shared__ float ws[32][68];    // 32 K x 64 N (padded stride 68)

  int mb = blockIdx.x;
  int ba = blockIdx.y;
  int b = ba / A_, a = ba % A_;
  int h = idx[b * A_ + a];
  int m0 = mb * 128;
  const float* xb = x + (size_t)b * S_ * D_;

  int tid = threadIdx.x;
  int wave = tid >> 5, lane = tid & 31;
  int lmod = lane & 15;
  int kh = 2 * (lane >> 4);

  v8f acc[4] = {};

  for (int k0 = 0; k0 < D_; k0 += 32) {
    // stage x tile (128x32) and W tile (32x64) via async DMA to LDS
    for (int i = tid; i < 1024; i += 256) {
      int row = i >> 3, c4 = i & 7;
      cp_b128(&xs[row][c4 * 4], xb + (size_t)(m0 + row) * D_ + k0 + c4 * 4);
    }
    for (int i = tid; i < 512; i += 256) {
      int row = i >> 4, c4 = i & 15;
      cp_b128(&ws[row][c4 * 4], W + (size_t)(k0 + row) * (H_ * DA_) + h * DA_ + c4 * 4);
    }
    cp_wait();
    __syncthreads();
    int mrow = 16 * wave + lmod;
    #pragma unroll
    for (int kk = 0; kk < 32; kk += 4) {
      v2f af;
      af.x = xs[mrow][kk + kh];
      af.y = xs[mrow][kk + kh + 1];
      #pragma unroll
      for (int j = 0; j < 4; ++j) {
        v2f bf;
        bf.x = ws[kk + kh][j * 16 + lmod];
        bf.y = ws[kk + kh + 1][j * 16 + lmod];
        acc[j] = wmma4(af, bf, acc[j]);
      }
    }
    __syncthreads();
  }

  // epilogue: + bias, store to (B, A, S, DA)
  float* ob = out + (((size_t)b * A_ + a) * S_) * DA_;
  int rbase = m0 + 16 * wave + 8 * (lane >> 4);
  #pragma unroll
  for (int j = 0; j < 4; ++j) {
    int col = j * 16 + lmod;
    float bvv = bias[h * DA_ + col];
    #pragma unroll
    for (int v = 0; v < 8; ++v) {
      int row = rbase + v;
      ob[(size_t)row * DA_ + col] = acc[j][v] + bvv;
    }
  }
}

// ---------------------------------------------------------------------------
// Kernel 4: flash attention per (b, a, q-tile of 64 rows).
// grid = (S/64, B*A), block = 256 (8 waves). Q frags in registers,
// K/V/S tiles in LDS (~52 KB). Online softmax in LDS (64 rows).
// ---------------------------------------------------------------------------
__global__ __launch_bounds__(256) void attn_kernel(
    const float* __restrict__ Qa, const float* __restrict__ Ka,
    const float* __restrict__ Va, const float* __restrict__ gate,
    float* __restrict__ Ofin) {
  __shared__ float Ks[64][68];
  __shared__ float Vs[64][64];
  __shared__ float Ss[64][68];
  __shared__ float mrow[64], lrow[64], arow[64];

  int qt = blockIdx.x;
  int ba = blockIdx.y;
  int b = ba / A_, a = ba % A_;
  size_t base = (((size_t)b * A_ + a) * S_) * DA_;
  const float* Q = Qa + base;
  const float* K = Ka + base;
  const float* V = Va + base;
  const float scale = 0.125f;   // 1/sqrt(64)

  int tid = threadIdx.x;
  int wave = tid >> 5, lane = tid & 31;
  int lmod = lane & 15;
  int kh = 2 * (lane >> 4);
  int ti = wave >> 1;           // row block (both tiles of this wave share it)

  // Preload this wave's Q fragments (rows 16*ti .. +15, all 64 K) into regs.
  v2f qf[16];
  {
    const float* qr = Q + (size_t)(qt * 64 + 16 * ti + lmod) * DA_;
    #pragma unroll
    for (int k4 = 0; k4 < 16; ++k4) {
      qf[k4].x = qr[k4 * 4 + kh] * scale;
      qf[k4].y = qr[k4 * 4 + kh + 1] * scale;
    }
  }
  if (tid < 64) { mrow[tid] = -3.0e38f; lrow[tid] = 0.0f; }

  v8f oacc[2] = {};

  for (int kt = 0; kt < S_ / 64; ++kt) {
    __syncthreads();   // protect Ks/Vs/Ss from previous iteration's readers
    // stage K and V tiles via async DMA to LDS (64x64 each)
    for (int i = tid; i < 1024; i += 256) {
      int row = i >> 4, c4 = i & 15;
      cp_b128(&Ks[row][c4 * 4], K + (size_t)(kt * 64 + row) * DA_ + c4 * 4);
      cp_b128(&Vs[row][c4 * 4], V + (size_t)(kt * 64 + row) * DA_ + c4 * 4);
    }
    cp_wait();
    __syncthreads();

    // S = (Q*scale) @ K^T : wave computes tiles (ti, tj) for tt=0,1
    #pragma unroll
    for (int tt = 0; tt < 2; ++tt) {
      int t = 2 * wave + tt;
      int tj = t & 3;
      int nr = 16 * tj + lmod;
      v8f s = {};
      #pragma unroll
      for (int k4 = 0; k4 < 16; ++k4) {
        v2f bf;
        bf.x = Ks[nr][k4 * 4 + kh];
        bf.y = Ks[nr][k4 * 4 + kh + 1];
        s = wmma4(qf[k4], bf, s);
      }
      #pragma unroll
      for (int v = 0; v < 8; ++v)
        Ss[16 * ti + v + 8 * (lane >> 4)][16 * tj + lmod] = s[v];
    }
    __syncthreads();

    // online softmax over the 64-wide slab (one thread per query row)
    if (tid < 64) {
      int r = tid;
      float mold = mrow[r];
      float mx = mold;
      for (int c = 0; c < 64; ++c) mx = fmaxf(mx, Ss[r][c]);
      float al = __expf(mold - mx);
      float sum = 0.0f;
      for (int c = 0; c < 64; ++c) {
        float p = __expf(Ss[r][c] - mx);
        Ss[r][c] = p;
        sum += p;
      }
      mrow[r] = mx;
      lrow[r] = lrow[r] * al + sum;
      arow[r] = al;
    }
    __syncthreads();

    // O = O*alpha + P @ V
    #pragma unroll
    for (int tt = 0; tt < 2; ++tt) {
      int t = 2 * wave + tt;
      int tj = t & 3;
      v8f o = oacc[tt];
      #pragma unroll
      for (int v = 0; v < 8; ++v)
        o[v] *= arow[16 * ti + v + 8 * (lane >> 4)];
      int mr = 16 * ti + lmod;
      #pragma unroll
      for (int k4 = 0; k4 < 16; ++k4) {
        v2f af;
        af.x = Ss[mr][k4 * 4 + kh];
        af.y = Ss[mr][k4 * 4 + kh + 1];
        v2f bf;
        bf.x = Vs[k4 * 4 + kh][16 * tj + lmod];
        bf.y = Vs[k4 * 4 + kh + 1][16 * tj + lmod];
        o = wmma4(af, bf, o);
      }
      oacc[tt] = o;
    }
  }

  // epilogue: O * gate / l  ->  Ofin (B, S, A*DA)
  float g = gate[b * A_ + a];
  #pragma unroll
  for (int tt = 0; tt < 2; ++tt) {
    int t = 2 * wave + tt;
    int tj = t & 3;
    int col = 16 * tj + lmod;
    #pragma unroll
    for (int v = 0; v < 8; ++v) {
      int row = 16 * ti + v + 8 * (lane >> 4);
      float val = oacc[tt][v] * g / lrow[row];
      Ofin[((size_t)b * S_ + qt * 64 + row) * (A_ * DA_) + a * DA_ + col] = val;
    }
  }
}

// ---------------------------------------------------------------------------
// Kernel 5: Z = Ofin(16384x256) @ Wo(256x1024) + bo -> d_out (B,S,D)
// grid = (16384/128, 1024/64), block = 256. Same tiling as kernel 3.
// ---------------------------------------------------------------------------
__global__ __launch_bounds__(256) void out_proj_kernel(
    const float* __restrict__ Ofin, const float* __restrict__ Wo,
    const float* __restrict__ bo, float* __restrict__ Z) {
  __shared__ float xs[128][36];
  __shared__ float ws[32][68];

  int m0 = blockIdx.x * 128;
  int n0 = blockIdx.y * 64;
  int tid = threadIdx.x;
  int wave = tid >> 5, lane = tid & 31;
  int lmod = lane & 15;
  int kh = 2 * (lane >> 4);
  const int KDIM = A_ * DA_;   // 256

  v8f acc[4] = {};

  for (int k0 = 0; k0 < KDIM; k0 += 32) {
    for (int i = tid; i < 1024; i += 256) {
      int row = i >> 3, c4 = i & 7;
      cp_b128(&xs[row][c4 * 4], Ofin + (size_t)(m0 + row) * KDIM + k0 + c4 * 4);
    }
    for (int i = tid; i < 512; i += 256) {
      int row = i >> 4, c4 = i & 15;
      cp_b128(&ws[row][c4 * 4], Wo + (size_t)(k0 + row) * D_ + n0 + c4 * 4);
    }
    cp_wait();
    __syncthreads();
    int mrow = 16 * wave + lmod;
    #pragma unroll
    for (int kk = 0; kk < 32; kk += 4) {
      v2f af;
      af.x = xs[mrow][kk + kh];
      af.y = xs[mrow][kk + kh + 1];
      #pragma unroll
      for (int j = 0; j < 4; ++j) {
        v2f bf;
        bf.x = ws[kk + kh][j * 16 + lmod];
        bf.y = ws[kk + kh + 1][j * 16 + lmod];
        acc[j] = wmma4(af, bf, acc[j]);
      }
    }
    __syncthreads();
  }

  int rbase = m0 + 16 * wave + 8 * (lane >> 4);
  #pragma unroll
  for (int j = 0; j < 4; ++j) {
    int col = n0 + j * 16 + lmod;
    float bvv = bo[col];
    #pragma unroll
    for (int v = 0; v < 8; ++v) {
      int row = rbase + v;
      Z[(size_t)row * D_ + col] = acc[j][v] + bvv;
    }
  }
}

// ---------------------------------------------------------------------------
extern "C" void kernel_launch(void* const* d_in, const int* in_sizes, int n_in,
                              void* d_out, int out_size, void* d_ws, size_t ws_size,
                              hipStream_t stream) {
  const float* x     = (const float*)d_in[0];
  const float* Wq    = (const float*)d_in[1];
  const float* bq    = (const float*)d_in[2];
  const float* Wk    = (const float*)d_in[3];
  const float* bk    = (const float*)d_in[4];
  const float* Wv    = (const float*)d_in[5];
  const float* bv    = (const float*)d_in[6];
  const float* Wr    = (const float*)d_in[7];
  const float* br    = (const float*)d_in[8];
  const float* gamma = (const float*)d_in[9];
  const float* beta  = (const float*)d_in[10];
  const float* Wo    = (const float*)d_in[11];
  const float* bo    = (const float*)d_in[12];
  // d_in[13] = n_active (fixed at 4 in these shapes)

  float* Z   = (float*)d_out;
  float* wsf = (float*)d_ws;

  float* xbar = wsf + XBAR_OFF;
  float* gate = wsf + GATE_OFF;
  int*   idx  = (int*)(wsf + IDX_OFF);
  float* Qa   = wsf + QACT_OFF;
  float* Ka   = wsf + KACT_OFF;
  float* Va   = wsf + VACT_OFF;
  float* Ofin = wsf + OFIN_OFF;

  mean_x_kernel<<<dim3(B_ * (D_ / 256)), dim3(256), 0, stream>>>(x, xbar);
  route_kernel<<<dim3(B_), dim3(256), 0, stream>>>(xbar, Wr, br, gamma, beta, idx, gate);
  qkv_proj_kernel<<<dim3(S_ / 128, B_ * A_, 3), dim3(256), 0, stream>>>(
      x, Wq, bq, Wk, bk, Wv, bv, idx, Qa, Ka, Va);
  attn_kernel<<<dim3(S_ / 64, B_ * A_), dim3(256), 0, stream>>>(Qa, Ka, Va, gate, Ofin);
  out_proj_kernel<<<dim3((B_ * S_) / 128, D_ / 64), dim3(256), 0, stream>>>(Ofin, Wo, bo, Z);
}